// Activation1d_65627100283695
// MI455X (gfx1250) — compile-verified
//
#include <hip/hip_runtime.h>

// ---------------------------------------------------------------------------
// Fused anti-aliased Snake activation (2x up-FIR -> snake -> 2x down-FIR)
// for MI455X (gfx1250). Single pass over HBM: ~268 MB traffic -> ~11.5us
// roofline at 23.3 TB/s. Down-conv = banded-Toeplitz matmul on the WMMA pipe
// (V_WMMA_F32_16X16X4_F32); up-conv + snake on VALU + v_sin_f32 (trans pipe).
// A-fragments come from a bank-conflict-free LDS Toeplitz table; the z buffer
// is skew-swizzled (phys = i + 2*(i>>5)) so B fetches hit 64 distinct banks.
// ---------------------------------------------------------------------------

typedef __attribute__((ext_vector_type(2))) float v2f;
typedef __attribute__((ext_vector_type(4))) float v4f;
typedef __attribute__((ext_vector_type(8))) float v8f;

#define B_SZ   8
#define C_SZ   512
#define T_SZ   8192
#define K_SZ   12
#define TILE   2048                 // outputs per workgroup (8 waves x 256)
#define NXP    (TILE + 16)          // staged x halo: xe[t0-8 .. t0+TILE+7]
#define NZ     (2*TILE + 11)        // z values needed: q in [2t0-5, 2t0+4101]
#define NZP    (2*TILE + 16)        // padded so WMMA's j=42,43 reads are defined
#define SZPHYS (NZP + 2*(NZP >> 5)) // skew-swizzled physical extent (4368)
#define ASTR   52                   // A-table row stride (bank-partitioning)
#define NTHR   256

__device__ __forceinline__ int zswz(int i) { return i + 2 * (i >> 5); }

__global__ __launch_bounds__(NTHR)
void snake_aa_kernel(const float* __restrict__ x,
                     const float* __restrict__ alpha,
                     const float* __restrict__ wu,
                     const float* __restrict__ wd,
                     float* __restrict__ out)
{
    __shared__ float sx[NXP];
    __shared__ float sz[SZPHYS];
    __shared__ float sA[16 * ASTR];

    const int tid = threadIdx.x;
    const int row = blockIdx.y;                 // b*C + c
    const int c   = row & (C_SZ - 1);
    const int t0  = blockIdx.x * TILE;
    const float* __restrict__ xr   = x   + (size_t)row * T_SZ;
    float*       __restrict__ orow = out + (size_t)row * T_SZ;

    // Up-filter taps -> registers (uniform addresses -> scalar loads)
    float wur[K_SZ];
#pragma unroll
    for (int k = 0; k < K_SZ; ++k) wur[k] = wu[k];

    // ---- Stage x halo: sx[i] = xe[t0-8+i]. Interior is always in-range ----
    for (int k = tid; k < TILE / 4; k += NTHR) {        // global_load_b128
        v4f v = *(const v4f*)(xr + t0 + 4 * k);
        *(v4f*)(&sx[8 + 4 * k]) = v;
    }
    if (tid < 16) {                                     // 16 clamped halo elems
        int i = (tid < 8) ? tid : (TILE + tid);
        int p = t0 - 8 + i;
        p = p < 0 ? 0 : (p > T_SZ - 1 ? T_SZ - 1 : p);
        sx[i] = xr[p];
    }

    // ---- Build Toeplitz A-table once: sA[m*ASTR + j] = wd[j-2m] (band) ----
    for (int e = tid; e < 16 * 64; e += NTHR) {
        const int mm = e >> 6, j = e & 63;
        if (j < 44) {
            const int idx = j - 2 * mm;
            float v = 0.0f;
            if ((unsigned)idx < (unsigned)K_SZ) v = wd[idx];
            sA[mm * ASTR + j] = v;
        }
    }
    __syncthreads();

    const float a     = __expf(alpha[c]);
    const float inv_a = 1.0f / (a + 1e-9f);

    // ---- Upsample (polyphase) + snake into swizzled LDS: sz <- ze[2t0-5+i] ----
    for (int i = tid; i < NZP; i += NTHR) {
        int qi = (i < NZ) ? i : (NZ - 1);        // pad region: repeat last valid
        int q  = 2 * t0 - 5 + qi;
        q = q < 0 ? 0 : (q > 2 * T_SZ - 1 ? 2 * T_SZ - 1 : q);
        const int s    = q >> 1;
        const int base = s - (t0 - 8);           // sx index of xe[s]
        float acc = 0.0f;
        if (q & 1) {                             // y[2s+1] = 2*sum_j wu[2j]*xe[s+3-j]
#pragma unroll
            for (int j = 0; j < 6; ++j)
                acc = __builtin_fmaf(wur[2 * j], sx[base + 3 - j], acc);
        } else {                                 // y[2s] = 2*sum_j wu[2j+1]*xe[s+2-j]
#pragma unroll
            for (int j = 0; j < 6; ++j)
                acc = __builtin_fmaf(wur[2 * j + 1], sx[base + 2 - j], acc);
        }
        const float y  = 2.0f * acc;
        const float sn = __sinf(a * y);          // v_sin_f32: trans pipe, co-issues
        sz[zswz(i)] = __builtin_fmaf(inv_a * sn, sn, y);
    }
    __syncthreads();

    // ---- Downsample FIR as banded matmul on the WMMA pipe ----
    // Per wave: 256 outputs, t = t0 + 256*w + 16*n + m.
    // D[m][n] = sum_j A[m][j]*B[j][n], j = 2m+k in [0,44),
    //   A[m][j] = wd[j-2m] (LDS table),  B[j][n] = sz[512w + 32n + j] (swizzled).
    const int lane = tid & 31;
    const int w    = tid >> 5;
    const int half = lane >> 4;                  // 0: K=4jb..+1, 1: K=4jb+2..+3
    const int m    = lane & 15;                  // A: M index / B,D: N index
    const int d0   = half * 2;

    const int abase = ASTR * m + d0;
    const int zb    = 512 * w + 32 * m + d0;
    v8f acc = {};
#pragma unroll
    for (int jb = 0; jb < 11; ++jb) {
        const int i0 = zb + 4 * jb;
        v2f av = *(const v2f*)(&sA[abase + 4 * jb]);   // ds_load_b64, 0-conflict
        v2f bv = *(const v2f*)(&sz[zswz(i0)]);         // ds_load_b64, 0-conflict
        acc = __builtin_amdgcn_wmma_f32_16x16x4_f32(
            /*neg_a=*/false, av, /*neg_b=*/false, bv,
            /*c_mod=*/(short)0, acc, /*reuse_a=*/false, /*reuse_b=*/false);
    }

    // D layout: lane<16 -> M=r, lane>=16 -> M=8+r; 8 contiguous outputs/lane
    const int tb = t0 + 256 * w + 16 * m + 8 * half;
    v4f lo = { acc[0], acc[1], acc[2], acc[3] };
    v4f hi = { acc[4], acc[5], acc[6], acc[7] };
    *(v4f*)(orow + tb)     = lo;                 // global_store_b128
    *(v4f*)(orow + tb + 4) = hi;
}

extern "C" void kernel_launch(void* const* d_in, const int* in_sizes, int n_in,
                              void* d_out, int out_size, void* d_ws, size_t ws_size,
                              hipStream_t stream)
{
    const float* x     = (const float*)d_in[0];   // (8, 512, 8192) f32
    const float* alpha = (const float*)d_in[1];   // (512,) f32
    const float* wu    = (const float*)d_in[2];   // (12,) f32
    const float* wd    = (const float*)d_in[3];   // (12,) f32
    float* out = (float*)d_out;                   // (8, 512, 8192) f32

    dim3 grid(T_SZ / TILE, B_SZ * C_SZ);          // (4, 4096)
    snake_aa_kernel<<<grid, NTHR, 0, stream>>>(x, alpha, wu, wd, out);
}